// VectorQuantizer_42167988912138
// MI455X (gfx1250) — compile-verified
//
#include <hip/hip_runtime.h>

typedef float v2f __attribute__((ext_vector_type(2)));
typedef float v8f __attribute__((ext_vector_type(8)));

#define KCODES 1024
#define DIM    64
#define HW     4096           // H*W
#define ROWPAD 68             // padded LDS row stride (floats) -> bank-conflict-free B loads

// LDS layout (floats):
//   [0, KCODES*ROWPAD)                 : codebook, padded rows
//   [E2_OFF, E2_OFF+KCODES)            : ||e||^2 per code
//   [IDX_OFF, IDX_OFF+128)             : per-token argmin index (int bits)
#define CB_FLOATS   (KCODES * ROWPAD)
#define E2_OFF      CB_FLOATS
#define IDX_OFF     (CB_FLOATS + KCODES)
#define SMEM_FLOATS (CB_FLOATS + KCODES + 128)

__global__ __launch_bounds__(256)
void vq_argmin_gather_kernel(const float* __restrict__ in,   // [16,64,64,64] NCHW
                             const float* __restrict__ cb,   // [1024,64]
                             float* __restrict__ out)        // [16,64,64,64] NCHW
{
    extern __shared__ float smem[];
    float* ldsCb  = smem;
    float* e2     = smem + E2_OFF;
    int*   idxLds = (int*)(smem + IDX_OFF);

    const int tid   = threadIdx.x;
    const int lane  = tid & 31;
    const int wv    = tid >> 5;     // wave id within block (0..7)
    const int m     = lane & 15;    // row/col within 16 (A: token M, B: code N)
    const int khalf = lane >> 4;    // which K-pair half this lane holds

    const int nb = blockIdx.x * 128;  // first flat token of this block (2 full h-rows)

    // ---- Phase 0: stage full codebook into LDS (padded rows), 16B granules
    {
        const float4* cb4 = (const float4*)cb;
        for (int i = tid; i < KCODES * (DIM / 4); i += 256) {
            int k = i >> 4;          // code row
            int j = i & 15;          // float4 within row
            float4 v = cb4[i];
            float* dst = &ldsCb[k * ROWPAD + j * 4];
            dst[0] = v.x; dst[1] = v.y; dst[2] = v.z; dst[3] = v.w;
        }
    }
    __syncthreads();

    // ---- Phase 0b: ||e||^2 per code
    for (int k = tid; k < KCODES; k += 256) {
        const float* row = &ldsCb[k * ROWPAD];
        float s = 0.f;
        #pragma unroll
        for (int d = 0; d < DIM; ++d) s += row[d] * row[d];
        e2[k] = s;
    }
    __syncthreads();

    // ---- Phase 1: each wave owns 16 tokens; WMMA f32 16x16x4 against all 1024 codes.
    // A-fragment layout (16x4 f32): lanes 0-15 hold M=lane, K={4c,4c+1}; lanes 16-31 hold K={4c+2,4c+3}.
    v2f a[16];
    {
        int tok = nb + wv * 16 + m;
        int w   = tok & 63;
        int hh  = (tok >> 6) & 63;
        int bb  = tok >> 12;
        const float* abase = in + (size_t)bb * DIM * HW + hh * 64 + w;
        #pragma unroll
        for (int c = 0; c < 16; ++c) {
            int kd = 4 * c + 2 * khalf;
            v2f t;
            t.x = abase[(size_t)kd * HW];
            t.y = abase[(size_t)(kd + 1) * HW];
            a[c] = t;
        }
    }

    float bestd[8];
    int   besti[8];
    #pragma unroll
    for (int r = 0; r < 8; ++r) { bestd[r] = 3.0e38f; besti[r] = 0; }

    for (int ct = 0; ct < KCODES / 16; ++ct) {
        v8f acc = {0.f, 0.f, 0.f, 0.f, 0.f, 0.f, 0.f, 0.f};
        // B-fragment (4x16 f32): VGPR j, lanes0-15 = row K=j (N=lane); lanes16-31 = row K=j+2.
        // Our lane reads code N=m, dims {4c+2*khalf, +1} -> consecutive floats -> ds_load_b64.
        const float* btile = &ldsCb[(ct * 16 + m) * ROWPAD + 2 * khalf];
        #pragma unroll
        for (int c = 0; c < 16; ++c) {
            v2f b = *(const v2f*)(btile + 4 * c);
            acc = __builtin_amdgcn_wmma_f32_16x16x4_f32(
                false, a[c], false, b, (short)0, acc, false, false);
        }
        // C layout: VGPR r -> (M=r, N=lane) lanes0-15, (M=r+8, N=lane-16) lanes16-31.
        float e2v  = e2[ct * 16 + m];
        int   code = ct * 16 + m;
        #pragma unroll
        for (int r = 0; r < 8; ++r) {
            float dist = __builtin_fmaf(-2.0f, acc[r], e2v);  // ||e||^2 - 2 x.e
            if (dist < bestd[r]) { bestd[r] = dist; besti[r] = code; }
        }
    }

    // min+index reduction across the 16 lanes of each half-wave (stays within halves)
    #pragma unroll
    for (int r = 0; r < 8; ++r) {
        float d  = bestd[r];
        int   bi = besti[r];
        #pragma unroll
        for (int off = 1; off <= 8; off <<= 1) {
            float od = __shfl_xor(d, off, 32);
            int   oi = __shfl_xor(bi, off, 32);
            if (od < d || (od == d && oi < bi)) { d = od; bi = oi; }
        }
        if (m == r) idxLds[wv * 16 + khalf * 8 + r] = bi;   // token M = r + 8*khalf
    }
    __syncthreads();

    // ---- Phase 2: gather codebook rows, write NCHW coalesced.
    // Block's 128 tokens = 2 full h-rows -> for fixed d, 128 consecutive floats.
    {
        int t     = tid & 127;
        int dbase = (tid >> 7) * 32;
        int tokG  = nb + t;
        int w     = tokG & 63;
        int hh    = (tokG >> 6) & 63;
        int bb    = tokG >> 12;
        int k     = idxLds[t];
        const float* crow = &ldsCb[k * ROWPAD];
        float* op = out + (size_t)bb * DIM * HW + hh * 64 + w;
        #pragma unroll
        for (int dd = 0; dd < 32; ++dd) {
            int d = dbase + dd;
            op[(size_t)d * HW] = crow[d];
        }
    }
}

extern "C" void kernel_launch(void* const* d_in, const int* in_sizes, int n_in,
                              void* d_out, int out_size, void* d_ws, size_t ws_size,
                              hipStream_t stream) {
    const float* in  = (const float*)d_in[0];   // [16,64,64,64] f32
    const float* cb  = (const float*)d_in[1];   // [1024,64] f32
    float*       out = (float*)d_out;           // [16,64,64,64] f32

    const size_t smemBytes = (size_t)SMEM_FLOATS * sizeof(float);  // ~277 KB (<= 320 KB/WGP)
    hipFuncSetAttribute((const void*)vq_argmin_gather_kernel,
                        hipFuncAttributeMaxDynamicSharedMemorySize, (int)smemBytes);

    const int nTokens = 16 * 64 * 64;           // 65536
    const int blocks  = nTokens / 128;          // 512
    vq_argmin_gather_kernel<<<blocks, 256, smemBytes, stream>>>(in, cb, out);
}